// OhemCrossEntropy_per_image_17729624998213
// MI455X (gfx1250) — compile-verified
//
#include <hip/hip_runtime.h>
#include <hip/hip_bf16.h>
#include <stdint.h>

// ---------------- problem constants (match reference) ----------------
#define BB        8
#define CC        19
#define HH        512
#define WW        1024
#define NN        (HH * WW)          // 524288 pixels per image
#define IGNORE_L  19
#define THRESH_F  0.8f
#define MIN_KEPT  100000

// ---------------- tiling / histogram config ----------------
#define NB        2048               // histogram bins over p in [0,1]
#define TILE      256                // pixels per TDM tile (19*256*4B = 19 KB LDS)
#define CHUNKS    64                 // blocks per image in pass1
#define ITERS     (NN / (CHUNKS * TILE))   // 32 tiles per block
#define P3BLK     64                 // pass3 blocks per image

typedef __attribute__((ext_vector_type(4))) unsigned int u32x4;
typedef __attribute__((ext_vector_type(8))) int          i32x8;
typedef __attribute__((ext_vector_type(4))) int          i32x4;

// ---------------------------------------------------------------------
// Issue one TDM 2-D tile load: TILE contiguous pixels x CC channels,
// channel stride = NN elements, f32 data. Descriptor per ISA 8.3/8.4.
// Must be executed by a single wave with uniform arguments.
// ---------------------------------------------------------------------
__device__ __forceinline__ void tdm_issue_tile(const float* score, unsigned lds_off,
                                               int b, int n0) {
  const unsigned long long ga =
      (unsigned long long)(size_t)score +
      ((unsigned long long)b * CC * NN + (unsigned long long)n0) * 4ull;
  // ---- D# group 0: count=1 | lds_addr | global_addr | type=2
  u32x4 g0;
  g0[0] = 1u;
  g0[1] = lds_off;
  g0[2] = (unsigned)(ga & 0xFFFFFFFFull);
  g0[3] = ((unsigned)(ga >> 32) & 0x01FFFFFFu) | (2u << 30);
  // ---- D# group 1: data_size=4B, tensor NN x CC, tile TILE x CC
  i32x8 g1;
  g1[0] = (int)(2u << 16);                                   // data_size code 2 (4B)
  g1[1] = (int)(((unsigned)NN & 0xFFFFu) << 16);             // tensor_dim0[15:0]
  g1[2] = (int)((((unsigned)NN >> 16) & 0xFFFFu) |
                (((unsigned)CC & 0xFFFFu) << 16));           // dim0 hi | dim1 lo
  g1[3] = (int)((((unsigned)CC >> 16) & 0xFFFFu) |
                ((unsigned)TILE << 16));                     // dim1 hi | tile_dim0
  g1[4] = (int)((unsigned)CC & 0xFFFFu);                     // tile_dim1=C, tile_dim2=0
  g1[5] = (int)(unsigned)NN;                                 // dim0_stride[31:0]
  g1[6] = 0;                                                 // stride0 hi | stride1 lo
  g1[7] = 0;
  const i32x4 gz4 = {0, 0, 0, 0};                            // groups 2/3 unused (2-D)
  const i32x8 gz8 = {0, 0, 0, 0, 0, 0, 0, 0};
  __builtin_amdgcn_tensor_load_to_lds(g0, g1, gz4, gz4, gz8, 0);
}

// =====================================================================
// pass1: double-buffered TDM staging of score tiles into LDS; compute
//        per-pixel loss & p_gt; per-image histogram via integer LDS
//        atomics (deterministic).
// grid = (CHUNKS, B), block = TILE threads (8 waves of 32)
// =====================================================================
__global__ __launch_bounds__(TILE) void ohem_pass1(
    const float* __restrict__ score, const int* __restrict__ target,
    float* __restrict__ pgt, float* __restrict__ lossb,
    int* __restrict__ hist, int* __restrict__ nvalid) {
  __shared__ float tileS[2][CC * TILE];  // ping-pong DMA buffers (38 KB)
  __shared__ int   lhist[NB];            // 8 KB
  __shared__ int   lvalid;

  const int b     = blockIdx.y;
  const int chunk = blockIdx.x;
  const int tid   = threadIdx.x;

  for (int i = tid; i < NB; i += TILE) lhist[i] = 0;
  if (tid == 0) lvalid = 0;
  __syncthreads();

  // LDS byte offsets of the two buffers (low 32 bits of flat shared addr).
  const unsigned lds0 = (unsigned)(size_t)(&tileS[0][0]);
  const unsigned lds1 = (unsigned)(size_t)(&tileS[1][0]);

  const int n_base = chunk * (ITERS * TILE);
  int myvalid = 0;

  if (tid < 32) tdm_issue_tile(score, lds0, b, n_base);   // prologue: tile 0

  for (int it = 0; it < ITERS; ++it) {
    const int cur = it & 1;
    if (tid < 32) {
      if (it + 1 < ITERS) {
        // issue next tile into the other buffer, then wait until only that
        // newest load is outstanding => tile `it` has landed (in-order TDM)
        tdm_issue_tile(score, (cur ? lds0 : lds1), b, n_base + (it + 1) * TILE);
        __builtin_amdgcn_s_wait_tensorcnt(1);
      } else {
        __builtin_amdgcn_s_wait_tensorcnt(0);
      }
    }
    __syncthreads();  // tile `it` visible to all waves

    const int  n     = n_base + it * TILE + tid;
    const int  t     = target[(size_t)b * NN + n];
    const bool valid = (t != IGNORE_L);
    const int  t0    = valid ? t : 0;

    float v[CC];
#pragma unroll
    for (int c = 0; c < CC; ++c) v[c] = tileS[cur][c * TILE + tid];

    float m = v[0];
#pragma unroll
    for (int c = 1; c < CC; ++c) m = fmaxf(m, v[c]);

    float s = 0.f, g = v[0];
#pragma unroll
    for (int c = 0; c < CC; ++c) {
      s += __expf(v[c] - m);
      if (c == t0) g = v[c];
    }
    const float ls = __logf(s);
    const float p  = __expf(g - m) * __builtin_amdgcn_rcpf(s);

    const size_t o = (size_t)b * NN + n;
    lossb[o] = valid ? (m - g + ls) : 0.f;       // -log softmax(gt)
    pgt[o]   = valid ? p : 2.0f;                 // 2.0 pad == reference sort pad

    if (valid) {
      int bin = (int)(p * (float)NB);
      bin = bin < 0 ? 0 : (bin > NB - 1 ? NB - 1 : bin);
      atomicAdd(&lhist[bin], 1);                 // ds_add (deterministic, int)
      ++myvalid;
    }
    // all waves done reading buffer `cur` before it is re-issued at it+2
    __syncthreads();
  }

  atomicAdd(&lvalid, myvalid);
  __syncthreads();
  if (tid == 0) atomicAdd(&nvalid[b], lvalid);
  for (int i = tid; i < NB; i += TILE) {
    const int h = lhist[i];
    if (h) atomicAdd(&hist[b * NB + i], h);
  }
}

// =====================================================================
// pass2: per image, locate k-th smallest p_gt bin -> OHEM threshold
// grid = B, block = 32 (scalar scan of 2048 bins)
// =====================================================================
__global__ __launch_bounds__(32) void ohem_pass2(
    const int* __restrict__ hist, const int* __restrict__ nvalid,
    float* __restrict__ thr) {
  const int b = blockIdx.x;
  if (threadIdx.x != 0) return;
  const int nv = nvalid[b];
  if (nv <= 0) { thr[b] = 2.0f; return; }       // matches p_sorted[-1] = 2.0 path
  const long long k = (long long)(MIN_KEPT < nv - 1 ? MIN_KEPT : nv - 1);
  const int* h = hist + b * NB;
  long long cum = 0;
  int bin = NB - 1;
  for (int i = 0; i < NB; ++i) {
    cum += h[i];
    if (cum >= k + 1) { bin = i; break; }
  }
  const float est = (float)(bin + 1) * (1.0f / (float)NB);  // kth <= est
  thr[b] = est > THRESH_F ? est : THRESH_F;     // exact whenever kth <= 0.8
}

// =====================================================================
// pass3: masked per-image sums with B128 loads; per-block partials
//        (no float atomics -> bitwise deterministic across replays)
// grid = (P3BLK, B), block = 256
// =====================================================================
__global__ __launch_bounds__(256) void ohem_pass3(
    const float* __restrict__ pgt, const float* __restrict__ lossb,
    const float* __restrict__ thr,
    float* __restrict__ part_loss, float* __restrict__ part_cnt) {
  const int   b    = blockIdx.y;
  const float t    = thr[b];
  const size_t base = (size_t)b * NN;
  const float4* pg4 = (const float4*)(pgt + base);
  const float4* lo4 = (const float4*)(lossb + base);
  const int nv4 = NN / 4;

  float sl = 0.f, sc = 0.f;
  for (int i = blockIdx.x * blockDim.x + threadIdx.x; i < nv4;
       i += gridDim.x * blockDim.x) {
    const float4 p = pg4[i];                     // invalid pixels carry 2.0
    const float4 l = lo4[i];
    if (p.x < t) { sl += l.x; sc += 1.f; }
    if (p.y < t) { sl += l.y; sc += 1.f; }
    if (p.z < t) { sl += l.z; sc += 1.f; }
    if (p.w < t) { sl += l.w; sc += 1.f; }
  }
  // wave32 shuffle reduction
  for (int off = 16; off > 0; off >>= 1) {
    sl += __shfl_down(sl, off, 32);
    sc += __shfl_down(sc, off, 32);
  }
  __shared__ float wsl[8], wsc[8];
  const int lane = threadIdx.x & 31, wv = threadIdx.x >> 5;
  if (lane == 0) { wsl[wv] = sl; wsc[wv] = sc; }
  __syncthreads();
  if (threadIdx.x == 0) {
    float a = 0.f, c = 0.f;
    for (int w = 0; w < 8; ++w) { a += wsl[w]; c += wsc[w]; }
    part_loss[b * P3BLK + blockIdx.x] = a;
    part_cnt[b * P3BLK + blockIdx.x]  = c;
  }
}

// =====================================================================
// finalize: fixed-order reduction -> mean of per-image means
// =====================================================================
__global__ void ohem_final(const float* __restrict__ part_loss,
                           const float* __restrict__ part_cnt,
                           float* __restrict__ out) {
  if (threadIdx.x != 0 || blockIdx.x != 0) return;
  float acc = 0.f;
  for (int b = 0; b < BB; ++b) {
    float sl = 0.f, sc = 0.f;
    for (int k = 0; k < P3BLK; ++k) {
      sl += part_loss[b * P3BLK + k];
      sc += part_cnt[b * P3BLK + k];
    }
    acc += sl / sc;                              // 0/0 -> NaN, like reference
  }
  out[0] = acc * (1.0f / (float)BB);
}

// =====================================================================
extern "C" void kernel_launch(void* const* d_in, const int* in_sizes, int n_in,
                              void* d_out, int out_size, void* d_ws, size_t ws_size,
                              hipStream_t stream) {
  (void)in_sizes; (void)n_in; (void)out_size; (void)ws_size;
  const float* score  = (const float*)d_in[0];
  const int*   target = (const int*)d_in[1];
  float*       out    = (float*)d_out;

  // ---- workspace layout ----
  char* ws = (char*)d_ws;
  const size_t sz_pn = (size_t)BB * NN * sizeof(float);
  float* pgt       = (float*)(ws);
  float* lossb     = (float*)(ws + sz_pn);
  int*   hist      = (int*)  (ws + 2 * sz_pn);
  int*   nvalid    = (int*)  (ws + 2 * sz_pn + (size_t)BB * NB * sizeof(int));
  float* thr       = (float*)(ws + 2 * sz_pn + ((size_t)BB * NB + BB) * sizeof(int));
  float* part_loss = (float*)(ws + 2 * sz_pn + ((size_t)BB * NB + 2 * BB) * sizeof(int));
  float* part_cnt  = part_loss + (size_t)BB * P3BLK;

  // zero only what is accumulated into (hist + nvalid); capture-legal
  (void)hipMemsetAsync(hist, 0, ((size_t)BB * NB + BB) * sizeof(int), stream);

  ohem_pass1<<<dim3(CHUNKS, BB), TILE, 0, stream>>>(score, target, pgt, lossb,
                                                    hist, nvalid);
  ohem_pass2<<<BB, 32, 0, stream>>>(hist, nvalid, thr);
  ohem_pass3<<<dim3(P3BLK, BB), 256, 0, stream>>>(pgt, lossb, thr,
                                                  part_loss, part_cnt);
  ohem_final<<<1, 32, 0, stream>>>(part_loss, part_cnt, out);
}